// MyGCN_89043261981497
// MI455X (gfx1250) — compile-verified
//
#include <hip/hip_runtime.h>
#include <math.h>

constexpr int kNodes = 100000;   // divisible by 16 (6250 tiles)
constexpr int kEdges = 3200000;

typedef __attribute__((ext_vector_type(2))) float v2f;
typedef __attribute__((ext_vector_type(8))) float v8f;

// ---------------------------------------------------------------- utilities
__global__ void k_zero(float* __restrict__ p, long long n) {
  long long i = (long long)blockIdx.x * blockDim.x + threadIdx.x;
  if (i < n) p[i] = 0.0f;
}

__global__ void k_deg_init(float* __restrict__ deg) {
  int i = blockIdx.x * blockDim.x + threadIdx.x;
  if (i < kNodes) deg[i] = 1.0f;  // self-loop contribution
}

__global__ void k_deg_accum(const int* __restrict__ dst, float* __restrict__ deg) {
  int e = blockIdx.x * blockDim.x + threadIdx.x;
  if (e < kEdges)
    __hip_atomic_fetch_add(&deg[dst[e]], 1.0f, __ATOMIC_RELAXED,
                           __HIP_MEMORY_SCOPE_AGENT);
}

__global__ void k_rsqrt_inplace(float* __restrict__ deg) {
  int i = blockIdx.x * blockDim.x + threadIdx.x;
  if (i < kNodes) deg[i] = rsqrtf(deg[i]);  // deg >= 1 always
}

// ------------------------------------------------- dense GEMM via f32 WMMA
// Y[kNodes x NOUT] = X[kNodes x K] * W[NOUT x K]^T
// One wave computes one 16x16 tile of Y with V_WMMA_F32_16X16X4_F32.
// A 16x4 f32 layout (ISA 7.12.2): lanes 0-15 -> M=lane, VGPR v = K=v;
// lanes 16-31 -> M=lane-16, VGPR v = K=v+2.  B mirrored (K over lanes/VGPRs,
// N = lane&15).  D: VGPR r -> row (lane<16 ? r : r+8), col = lane&15.
template <int K, int NOUT>
__global__ void k_gemm_wmma(const float* __restrict__ X,
                            const float* __restrict__ W,
                            float* __restrict__ Y) {
  constexpr int kTilesN = NOUT / 16;
  constexpr int kTiles  = (kNodes / 16) * kTilesN;
  int wave = (int)((blockIdx.x * (long long)blockDim.x + threadIdx.x) >> 5);
  if (wave >= kTiles) return;          // wave-uniform: EXEC stays all-ones
  int lane = threadIdx.x & 31;
  int tm = wave / kTilesN;
  int tn = wave % kTilesN;
  int row = tm * 16 + (lane & 15);
  int col = tn * 16 + (lane & 15);
  int kh  = (lane >> 4) * 2;           // 0 for lanes 0-15, 2 for 16-31
  const float* xr = X + (size_t)row * K;
  const float* wr = W + (size_t)col * K;
  v8f c = {0.f, 0.f, 0.f, 0.f, 0.f, 0.f, 0.f, 0.f};
#pragma unroll
  for (int k0 = 0; k0 < K; k0 += 4) {
    v2f a, b;
    a[0] = xr[k0 + kh + 0];
    a[1] = xr[k0 + kh + 1];
    b[0] = wr[k0 + kh + 0];
    b[1] = wr[k0 + kh + 1];
    c = __builtin_amdgcn_wmma_f32_16x16x4_f32(false, a, false, b,
                                              (short)0, c, false, false);
  }
  int mbase = tm * 16 + (lane >> 4) * 8;
  int n     = tn * 16 + (lane & 15);
#pragma unroll
  for (int r = 0; r < 8; ++r)
    Y[(size_t)(mbase + r) * NOUT + n] = c[r];
}

// ------------------------------------------- edge scatter (message passing)
// 32 (or 16) consecutive lanes handle one edge: src/dst/norm are uniform within each edge's lane group,
// gather and atomic-scatter are coalesced 128B/64B lines held in L2.
template <int F, int LOGF>
__global__ void k_scatter(const int* __restrict__ src, const int* __restrict__ dst,
                          const float* __restrict__ dinv,
                          const float* __restrict__ hin, float* __restrict__ acc) {
  long long gid = (long long)blockIdx.x * blockDim.x + threadIdx.x;
  long long e = gid >> LOGF;
  int f = (int)(gid & (F - 1));
  if (e >= kEdges) return;
  int s = src[e], d = dst[e];
  float norm = dinv[s] * dinv[d];
  float v = hin[(size_t)s * F + f] * norm;
  __hip_atomic_fetch_add(&acc[(size_t)d * F + f], v, __ATOMIC_RELAXED,
                         __HIP_MEMORY_SCOPE_AGENT);
}

// self-loop term + bias (+ optional ReLU)
template <int F, bool RELU>
__global__ void k_finalize(const float* __restrict__ acc,
                           const float* __restrict__ hlin,
                           const float* __restrict__ dinv,
                           const float* __restrict__ bias,
                           float* __restrict__ out) {
  long long gid = (long long)blockIdx.x * blockDim.x + threadIdx.x;
  if (gid >= (long long)kNodes * F) return;
  int v = (int)(gid / F);
  int f = (int)(gid % F);
  float di = dinv[v];
  float val = acc[gid] + hlin[gid] * (di * di) + bias[f];
  out[gid] = RELU ? fmaxf(val, 0.0f) : val;
}

// ------------------------------------------------------------ edge decoder
__global__ void k_decode(const int* __restrict__ src, const int* __restrict__ dst,
                         const float* __restrict__ z, float* __restrict__ out) {
  int e = blockIdx.x * blockDim.x + threadIdx.x;
  if (e >= kEdges) return;
  const float4* zs = (const float4*)(z + (size_t)src[e] * 16);
  const float4* zd = (const float4*)(z + (size_t)dst[e] * 16);
  float acc = 0.f;
#pragma unroll
  for (int r = 0; r < 4; ++r) {
    float4 a = zs[r], b = zd[r];
    acc += a.x * b.x + a.y * b.y + a.z * b.z + a.w * b.w;
  }
  out[e] = 1.0f / (1.0f + expf(-acc));
}

// ------------------------------------------------------------------ driver
extern "C" void kernel_launch(void* const* d_in, const int* in_sizes, int n_in,
                              void* d_out, int out_size, void* d_ws, size_t ws_size,
                              hipStream_t stream) {
  const float* x  = (const float*)d_in[0];
  const int*   ei = (const int*)d_in[1];   // [2, E] int32 (x64 disabled in JAX)
  const float* W1 = (const float*)d_in[2]; // [32,64]
  const float* b1 = (const float*)d_in[3];
  const float* W2 = (const float*)d_in[4]; // [16,32]
  const float* b2 = (const float*)d_in[5];
  const int* src = ei;
  const int* dst = ei + kEdges;
  float* out = (float*)d_out;

  // workspace layout (floats); zlin/acc2 alias dead layer-1 regions
  float* ws   = (float*)d_ws;
  float* dinv = ws;                               // kNodes  (deg then rsqrt)
  float* hlin = ws + 102400;                      // kNodes*32
  float* acc1 = hlin + (size_t)kNodes * 32;       // kNodes*32
  float* h    = acc1 + (size_t)kNodes * 32;       // kNodes*32
  float* zlin = hlin;                             // reuse: kNodes*16
  float* acc2 = acc1;                             // reuse: kNodes*16
  float* z    = h + (size_t)kNodes * 32;          // kNodes*16
  (void)in_sizes; (void)n_in; (void)out_size; (void)ws_size;

  const int B = 256;
  auto nb = [](long long n, int b) { return (int)((n + b - 1) / b); };

  // degrees + dinv
  k_deg_init<<<nb(kNodes, B), B, 0, stream>>>(dinv);
  k_zero<<<nb((long long)kNodes * 32, B), B, 0, stream>>>(acc1, (long long)kNodes * 32);
  k_deg_accum<<<nb(kEdges, B), B, 0, stream>>>(dst, dinv);
  k_rsqrt_inplace<<<nb(kNodes, B), B, 0, stream>>>(dinv);

  // layer 1: hlin = x @ W1^T ; aggregate ; +self-loop +bias ; ReLU
  k_gemm_wmma<64, 32><<<nb((long long)(kNodes / 16) * 2 * 32, B), B, 0, stream>>>(x, W1, hlin);
  k_scatter<32, 5><<<nb((long long)kEdges * 32, B), B, 0, stream>>>(src, dst, dinv, hlin, acc1);
  k_finalize<32, true><<<nb((long long)kNodes * 32, B), B, 0, stream>>>(acc1, hlin, dinv, b1, h);

  // layer 2
  k_zero<<<nb((long long)kNodes * 16, B), B, 0, stream>>>(acc2, (long long)kNodes * 16);
  k_gemm_wmma<32, 16><<<nb((long long)(kNodes / 16) * 1 * 32, B), B, 0, stream>>>(h, W2, zlin);
  k_scatter<16, 4><<<nb((long long)kEdges * 16, B), B, 0, stream>>>(src, dst, dinv, zlin, acc2);
  k_finalize<16, false><<<nb((long long)kNodes * 16, B), B, 0, stream>>>(acc2, zlin, dinv, b2, z);

  // decode: sigmoid(dot(z[src], z[dst]))
  k_decode<<<nb(kEdges, B), B, 0, stream>>>(src, dst, z, out);
}